// ChamferLoss_33861522161768
// MI455X (gfx1250) — compile-verified
//
#include <hip/hip_runtime.h>

typedef __attribute__((ext_vector_type(2))) float v2f;
typedef __attribute__((ext_vector_type(8))) float v8f;

#define BATCH  64
#define NPTS   2048
#define NTILES (NPTS / 16)   // 128

// ---- monotonic float<->uint key (total order, min-safe for negatives) ----
__device__ __forceinline__ unsigned f2key(float f) {
    unsigned u = __float_as_uint(f);
    return (u & 0x80000000u) ? ~u : (u | 0x80000000u);
}
__device__ __forceinline__ float key2f(unsigned k) {
    return (k & 0x80000000u) ? __uint_as_float(k & 0x7FFFFFFFu)
                             : __uint_as_float(~k);
}

__global__ __launch_bounds__(256) void init_kernel(unsigned* keys, float* out, int total) {
    int i = blockIdx.x * blockDim.x + threadIdx.x;
    if (i < total) keys[i] = 0xFFFFFFFFu;   // +infinity-like key
    if (i == 0)    out[0] = 0.0f;
}

// One block (4 waves) per (batch, p row-tile). Each wave sweeps 32 q col-tiles,
// one V_WMMA_F32_16X16X4_F32 per 16x16 distance tile. Inner loop has no DS ops
// and no EXEC divergence; col-min halves are merged by the atomic itself.
__global__ __launch_bounds__(128)
void chamfer_kernel(const float* __restrict__ p, const float* __restrict__ q,
                    unsigned* __restrict__ rowkeys,   // B*N: min over m per p_n
                    unsigned* __restrict__ colkeys)   // B*N: min over n per q_m
{
    const int b     = blockIdx.x / NTILES;
    const int itile = blockIdx.x % NTILES;
    const int lane  = threadIdx.x & 31;
    const int wave  = threadIdx.x >> 5;

    const int m = lane & 15;   // tile row (p) / tile col (q) this lane owns
    const int h = lane >> 4;   // K-half (WMMA 32-bit A/B layout)

    // ---- A tile: 16 p rows x K=4, scaled by -2 so WMMA yields -2*p.q ----
    const float* pbase = p + ((size_t)b * NPTS + itile * 16 + m) * 4;
    float4 p4 = *(const float4*)pbase;                      // full row, lane-local |p|^2
    float psq = p4.x * p4.x + p4.y * p4.y + p4.z * p4.z + p4.w * p4.w;
    v2f a2;                                                 // this lane's K-half, *-2
    a2.x = -2.0f * (h ? p4.z : p4.x);
    a2.y = -2.0f * (h ? p4.w : p4.y);

    // |p|^2 for the 8 rows this lane's C registers map to: row = r + 8*h
    float ps[8];
#pragma unroll
    for (int r = 0; r < 8; ++r)
        ps[r] = __shfl(psq, r + 8 * h, 32);                 // one-time bpermutes

    float rowacc[8];
#pragma unroll
    for (int r = 0; r < 8; ++r) rowacc[r] = __builtin_inff();

    const float* qb = q + (size_t)b * NPTS * 4;
    unsigned* colbase = colkeys + (size_t)b * NPTS;

#pragma unroll 4
    for (int t = 0; t < NTILES / 4; ++t) {
        const int j = 4 * t + wave;
        const float* qbase = qb + (j * 16 + m) * 4;
        // B tile operand (this lane's K-half) + full column for lane-local |q|^2
        v2f    bm = *(const v2f*)(qbase + 2 * h);
        float4 q4 = *(const float4*)qbase;
        float qsq = q4.x * q4.x + q4.y * q4.y + q4.z * q4.z + q4.w * q4.w;

        v8f c = {};
        // c = (-2p) x q : one 16x16 tile of cross terms
        c = __builtin_amdgcn_wmma_f32_16x16x4_f32(
                /*neg_a=*/false, a2, /*neg_b=*/false, bm,
                /*c_mod=*/(short)0, c, /*reuse_a=*/false, /*reuse_b=*/false);

        float colcand = __builtin_inff();
#pragma unroll
        for (int r = 0; r < 8; ++r) {
            rowacc[r] = fminf(rowacc[r], qsq + c[r]);    // + |q_m|^2 (row-min path)
            colcand   = fminf(colcand,   ps[r] + c[r]);  // + |p_n|^2 (col-min path)
        }
        // both halves hit the same address: atomic merges the 8+8 row partials
        atomicMin(&colbase[j * 16 + m], f2key(qsq + colcand));
    }

    // reduce row mins across the 16 column-lanes of each half, then merge globally
#pragma unroll
    for (int r = 0; r < 8; ++r) {
        float v = rowacc[r];
        v = fminf(v, __shfl_xor(v, 1, 32));
        v = fminf(v, __shfl_xor(v, 2, 32));
        v = fminf(v, __shfl_xor(v, 4, 32));
        v = fminf(v, __shfl_xor(v, 8, 32));
        if (m == 0) {
            float rowval = ps[r] + v;                    // + |p_row|^2, row = r + 8*h
            atomicMin(&rowkeys[(size_t)b * NPTS + itile * 16 + r + 8 * h], f2key(rowval));
        }
    }
}

// jet term: | sum_n p[b,n,:] - sum_n q[b,n,:] |^2, one block per batch
__global__ __launch_bounds__(256)
void jet_kernel(const float* __restrict__ p, const float* __restrict__ q, float* out) {
    const int b = blockIdx.x, tid = threadIdx.x;
    const float4* pp = (const float4*)(p + (size_t)b * NPTS * 4);
    const float4* qq = (const float4*)(q + (size_t)b * NPTS * 4);
    float4 acc = make_float4(0.f, 0.f, 0.f, 0.f);
    for (int n = tid; n < NPTS; n += 256) {
        float4 pv = pp[n], qv = qq[n];
        acc.x += pv.x - qv.x; acc.y += pv.y - qv.y;
        acc.z += pv.z - qv.z; acc.w += pv.w - qv.w;
    }
#pragma unroll
    for (int s = 1; s < 32; s <<= 1) {
        acc.x += __shfl_xor(acc.x, s, 32);
        acc.y += __shfl_xor(acc.y, s, 32);
        acc.z += __shfl_xor(acc.z, s, 32);
        acc.w += __shfl_xor(acc.w, s, 32);
    }
    __shared__ float4 lds[8];
    int wave = tid >> 5, lane = tid & 31;
    if (lane == 0) lds[wave] = acc;
    __syncthreads();
    if (tid == 0) {
        float4 t = make_float4(0.f, 0.f, 0.f, 0.f);
        for (int w = 0; w < 8; ++w) {
            t.x += lds[w].x; t.y += lds[w].y; t.z += lds[w].z; t.w += lds[w].w;
        }
        atomicAdd(out, t.x * t.x + t.y * t.y + t.z * t.z + t.w * t.w);
    }
}

// sum all row/col mins (keys -> float) into the scalar output
__global__ __launch_bounds__(256)
void reduce_kernel(const unsigned* __restrict__ keys, float* out, int total) {
    int i = blockIdx.x * blockDim.x + threadIdx.x;
    float v = (i < total) ? key2f(keys[i]) : 0.0f;
#pragma unroll
    for (int s = 1; s < 32; s <<= 1) v += __shfl_xor(v, s, 32);
    __shared__ float lds[8];
    int wave = threadIdx.x >> 5, lane = threadIdx.x & 31;
    if (lane == 0) lds[wave] = v;
    __syncthreads();
    if (threadIdx.x == 0) {
        float t = 0.f;
        for (int w = 0; w < 8; ++w) t += lds[w];
        atomicAdd(out, t);
    }
}

extern "C" void kernel_launch(void* const* d_in, const int* in_sizes, int n_in,
                              void* d_out, int out_size, void* d_ws, size_t ws_size,
                              hipStream_t stream) {
    const float* p = (const float*)d_in[0];
    const float* q = (const float*)d_in[1];
    float* out = (float*)d_out;

    unsigned* rowkeys = (unsigned*)d_ws;            // B*N keys
    unsigned* colkeys = rowkeys + BATCH * NPTS;     // B*N keys (adjacent)
    const int total = 2 * BATCH * NPTS;             // 1 MiB of d_ws

    init_kernel<<<(total + 255) / 256, 256, 0, stream>>>(rowkeys, out, total);
    chamfer_kernel<<<BATCH * NTILES, 128, 0, stream>>>(p, q, rowkeys, colkeys);
    jet_kernel<<<BATCH, 256, 0, stream>>>(p, q, out);
    reduce_kernel<<<(total + 255) / 256, 256, 0, stream>>>(rowkeys, out, total);
}